// FinePreprocess_52939766891089
// MI455X (gfx1250) — compile-verified
//
#include <hip/hip_runtime.h>

// ---- compile-time probes (show up in stderr; no codegen effect) ----------
#if __has_builtin(__builtin_amdgcn_global_load_async_to_lds_b32)
#warning "PROBE: __builtin_amdgcn_global_load_async_to_lds_b32 AVAILABLE"
#endif
#if __has_builtin(__builtin_amdgcn_global_load_async_to_lds_b128)
#warning "PROBE: __builtin_amdgcn_global_load_async_to_lds_b128 AVAILABLE"
#endif
#if __has_builtin(__builtin_amdgcn_global_store_async_from_lds_b32)
#warning "PROBE: __builtin_amdgcn_global_store_async_from_lds_b32 AVAILABLE"
#endif
#if __has_builtin(__builtin_amdgcn_global_store_async_from_lds_b128)
#warning "PROBE: __builtin_amdgcn_global_store_async_from_lds_b128 AVAILABLE"
#endif
#if __has_builtin(__builtin_amdgcn_s_wait_asynccnt)
#warning "PROBE: __builtin_amdgcn_s_wait_asynccnt AVAILABLE"
#endif
#if __has_builtin(__builtin_amdgcn_ds_atomic_async_barrier_arrive_b64)
#warning "PROBE: __builtin_amdgcn_ds_atomic_async_barrier_arrive_b64 AVAILABLE"
#endif

// ---- problem constants (fixed by reference setup) ------------------------
constexpr int kC      = 128;
constexpr int kH      = 240;
constexpr int kW      = 320;
constexpr int kWK     = 5;
constexpr int kPAD    = 2;
constexpr int kSTRIDE = 4;
constexpr int kOUTW   = (kW + 2 * kPAD - kWK) / kSTRIDE + 1;   // 80
constexpr int kPATCH  = kC * kWK * kWK;                        // 3200 (multiple of 32)

// out[m*3200 + f] = feat[b, f/25, gy*4 + (f%25)/5 - 2, gx*4 + f%5 - 2], 0 if OOB.
// feat1 half is appended after the feat0 half (tuple concat, both M*3200 floats).
__global__ __launch_bounds__(256)
void FinePreprocess_gather_kernel(const float* __restrict__ feat0,
                                  const float* __restrict__ feat1,
                                  const int*   __restrict__ b_ids,
                                  const int*   __restrict__ i_ids,
                                  const int*   __restrict__ j_ids,
                                  float*       __restrict__ out,
                                  unsigned M, unsigned total)
{
    const unsigned o = blockIdx.x * blockDim.x + threadIdx.x;
    if (o >= total) return;

    const unsigned half   = M * (unsigned)kPATCH;      // 19,200,000: multiple of 32
    const bool     second = (o >= half);               // wave-uniform
    const unsigned oo     = second ? (o - half) : o;

    // kPATCH = 3200 = 100*32  =>  every wave sees exactly one match index m.
    unsigned m = oo / (unsigned)kPATCH;
    m = (unsigned)__builtin_amdgcn_readfirstlane((int)m);   // force scalar idx loads
    const unsigned f  = oo - m * (unsigned)kPATCH;          // 0..3199, per-lane

    const unsigned ch = f / 25u;
    const unsigned r  = f - ch * 25u;
    const unsigned ky = r / 5u;
    const unsigned kx = r - ky * 5u;

    const int b   = b_ids[m];
    const int idx = second ? j_ids[m] : i_ids[m];
    const int gy  = idx / kOUTW;
    const int gx  = idx - gy * kOUTW;

    const int y = gy * kSTRIDE + (int)ky - kPAD;   // [-2, 238]
    const int x = gx * kSTRIDE + (int)kx - kPAD;   // [-2, 318]

    float v = 0.0f;
    if (((unsigned)y < (unsigned)kH) & ((unsigned)x < (unsigned)kW)) {
        const float* __restrict__ src = second ? feat1 : feat0;
        const size_t off = (((size_t)b * kC + ch) * kH + (size_t)y) * kW + (size_t)x;
        v = src[off];
    }

    // Write-once streaming output: non-temporal so the 154MB of stores do not
    // evict gather-source lines from the 192MB L2.
    __builtin_nontemporal_store(v, out + o);
}

extern "C" void kernel_launch(void* const* d_in, const int* in_sizes, int n_in,
                              void* d_out, int out_size, void* d_ws, size_t ws_size,
                              hipStream_t stream) {
    (void)n_in; (void)d_ws; (void)ws_size;
    // setup_inputs order: feat_f0, feat_f1, hw0_f, hw0_c, b_ids, i_ids, j_ids
    const float* feat0 = (const float*)d_in[0];
    const float* feat1 = (const float*)d_in[1];
    const int*   b_ids = (const int*)d_in[4];
    const int*   i_ids = (const int*)d_in[5];
    const int*   j_ids = (const int*)d_in[6];
    float*       out   = (float*)d_out;

    const unsigned M     = (unsigned)in_sizes[4];      // 6000
    const unsigned total = (unsigned)out_size;         // 2*M*3200 = 38,400,000

    const int block = 256;
    const int grid  = (int)((total + (unsigned)block - 1u) / (unsigned)block);  // 150000
    FinePreprocess_gather_kernel<<<grid, block, 0, stream>>>(
        feat0, feat1, b_ids, i_ids, j_ids, out, M, total);
}